// HierarchyLossWithSegments_13142599926432
// MI455X (gfx1250) — compile-verified
//
#include <hip/hip_runtime.h>
#include <hip/hip_bf16.h>
#include <math.h>

#define NUM_CLASSES 1024
#define BLOCK_THREADS 256

typedef __attribute__((ext_vector_type(2))) float v2f;
typedef __attribute__((ext_vector_type(8))) float v8f;

// Exact f32 sum of all 32 lanes' `p` using V_WMMA_F32_16X16X4_F32.
// A (16x4): lane<16 -> A[lane][0]=p, A[lane][1]=0 ; lane>=16 -> A[lane-16][2]=p, A[lane-16][3]=0
// B (4x16): all ones  =>  D[m][n] = p(m) + p(m+16) for every n.
// Lane n<16 holds D[0..7][n] in its 8 dest VGPRs, lane n>=16 holds D[8..15][n-16].
// sum(own 8) + shfl_xor(16) == full wave sum, identical on all lanes.
__device__ __forceinline__ float wave_sum_wmma(float p) {
    v2f a; a[0] = p;    a[1] = 0.0f;
    v2f b; b[0] = 1.0f; b[1] = 1.0f;
    v8f c = {};
    v8f d = __builtin_amdgcn_wmma_f32_16x16x4_f32(
        /*neg_a=*/false, a, /*neg_b=*/false, b,
        /*c_mod=*/(short)0, c, /*reuse_a=*/false, /*reuse_b=*/false);
    float s = ((d[0] + d[1]) + (d[2] + d[3])) + ((d[4] + d[5]) + (d[6] + d[7]));
    s += __shfl_xor(s, 16, 32);
    return s;
}

__device__ __forceinline__ float4 max4(float4 a, float4 b) {
    float4 r;
    r.x = fmaxf(a.x, b.x); r.y = fmaxf(a.y, b.y);
    r.z = fmaxf(a.z, b.z); r.w = fmaxf(a.w, b.w);
    return r;
}

// y*log(p1) + (1-y)*log1p(-p1) + y*log(p2) + (1-y)*log1p(-p2)
__device__ __forceinline__ float bce2(float p1, float p2, float y) {
    return y * (logf(p1) + logf(p2)) + (1.0f - y) * (log1pf(-p1) + log1pf(-p2));
}

// One block per video: ragged column-max over that video's sections, then the
// BCE log-terms for both score tensors, reduced to one partial per block.
__global__ void __launch_bounds__(BLOCK_THREADS)
segmax_bce_kernel(const float* __restrict__ sec,    // [T, C]
                  const float* __restrict__ vid,    // [B, C]
                  const float* __restrict__ lab,    // [B, C]
                  const int*   __restrict__ segids, // [T] sorted
                  float*       __restrict__ block_sums,
                  int total_sections) {
    const int b   = blockIdx.x;
    const int tid = threadIdx.x;

    // Binary search segment bounds (uniform across the block -> no divergence).
    int lo = 0, hi = total_sections;
    while (lo < hi) { int mid = (lo + hi) >> 1; if (segids[mid] <  b)     lo = mid + 1; else hi = mid; }
    const int start = lo;
    hi = total_sections;
    while (lo < hi) { int mid = (lo + hi) >> 1; if (segids[mid] <= b)     lo = mid + 1; else hi = mid; }
    const int nrows = lo - start;

    // Column-wise max over this video's rows; float4 per thread (coalesced b128),
    // 4 independent accumulators for memory-level parallelism.
    const int stride4 = NUM_CLASSES / 4;
    const float4* base = reinterpret_cast<const float4*>(sec) + (size_t)start * stride4 + tid;
    const float NEG_INF = -__builtin_inff();
    float4 m0 = {NEG_INF, NEG_INF, NEG_INF, NEG_INF}, m1 = m0, m2 = m0, m3 = m0;
    int r = 0;
    for (; r + 4 <= nrows; r += 4) {
        float4 a0 = base[(size_t)(r + 0) * stride4];
        float4 a1 = base[(size_t)(r + 1) * stride4];
        float4 a2 = base[(size_t)(r + 2) * stride4];
        float4 a3 = base[(size_t)(r + 3) * stride4];
        m0 = max4(m0, a0); m1 = max4(m1, a1);
        m2 = max4(m2, a2); m3 = max4(m3, a3);
    }
    for (; r < nrows; ++r) m0 = max4(m0, base[(size_t)r * stride4]);
    float4 m = max4(max4(m0, m1), max4(m2, m3));

    // BCE log-terms for both tensors at this (video, 4 classes).
    const size_t vo = (size_t)b * NUM_CLASSES + (size_t)tid * 4;
    const float4 p1 = *reinterpret_cast<const float4*>(vid + vo);
    const float4 y  = *reinterpret_cast<const float4*>(lab + vo);
    float s = bce2(p1.x, m.x, y.x) + bce2(p1.y, m.y, y.y)
            + bce2(p1.z, m.z, y.z) + bce2(p1.w, m.w, y.w);

    // Wave reduce (WMMA, exact f32), then combine the 8 waves via LDS.
    float wsum = wave_sum_wmma(s);
    __shared__ float lds[BLOCK_THREADS / 32];
    if ((tid & 31) == 0) lds[tid >> 5] = wsum;
    __syncthreads();
    if (tid == 0) {
        float t = 0.0f;
        #pragma unroll
        for (int i = 0; i < BLOCK_THREADS / 32; ++i) t += lds[i];
        block_sums[b] = t;
    }
}

// Single-block deterministic final sum (fixed order; no float atomics anywhere).
__global__ void __launch_bounds__(BLOCK_THREADS)
finalize_kernel(const float* __restrict__ block_sums, int n,
                float* __restrict__ out, float inv_den) {
    const int tid = threadIdx.x;
    float s = 0.0f;
    for (int i = tid; i < n; i += BLOCK_THREADS) s += block_sums[i];
    float wsum = wave_sum_wmma(s);
    __shared__ float lds[BLOCK_THREADS / 32];
    if ((tid & 31) == 0) lds[tid >> 5] = wsum;
    __syncthreads();
    if (tid == 0) {
        float t = 0.0f;
        #pragma unroll
        for (int i = 0; i < BLOCK_THREADS / 32; ++i) t += lds[i];
        out[0] = -t * inv_den;   // -(S1+S2)/(B*C)
    }
}

extern "C" void kernel_launch(void* const* d_in, const int* in_sizes, int n_in,
                              void* d_out, int out_size, void* d_ws, size_t ws_size,
                              hipStream_t stream) {
    const float* sec    = (const float*)d_in[0];   // section_scores [T, C]
    const float* vid    = (const float*)d_in[1];   // video_scores   [B, C]
    const float* lab    = (const float*)d_in[2];   // labels         [B, C]
    const int*   segids = (const int*)  d_in[3];   // segment_ids    [T]

    const int total_sections = in_sizes[0] / NUM_CLASSES;
    const int num_videos     = in_sizes[1] / NUM_CLASSES;

    float* block_sums = (float*)d_ws;              // num_videos floats
    float* out        = (float*)d_out;

    segmax_bce_kernel<<<num_videos, BLOCK_THREADS, 0, stream>>>(
        sec, vid, lab, segids, block_sums, total_sections);

    const float inv_den = 1.0f / ((float)num_videos * (float)NUM_CLASSES);
    finalize_kernel<<<1, BLOCK_THREADS, 0, stream>>>(block_sums, num_videos, out, inv_den);
}